// AttentionLayer_36687610642850
// MI455X (gfx1250) — compile-verified
//
#include <hip/hip_runtime.h>

// Shapes fixed by the reference harness:
//   b=4, t_q=128, t_k=512, n_q=n_k=512, out=512
#define BATCH   4
#define TQ      128
#define TK      512
#define NDIM    512
#define NOUT    512
#define NROWS   (BATCH * TQ)      // 512 flattened (b,q) rows
#define CATDIM  (NDIM + NDIM)     // 1024

typedef float v2f __attribute__((ext_vector_type(2)));
typedef float v8f __attribute__((ext_vector_type(8)));

// ---------- fast transcendentals (hardware trans ops, no ocml calls) -------
__device__ __forceinline__ float fast_exp2(float x) {
#if __has_builtin(__builtin_amdgcn_exp2f)
  return __builtin_amdgcn_exp2f(x);
#else
  return exp2f(x);
#endif
}

__device__ __forceinline__ float fast_rcp(float x) {
#if __has_builtin(__builtin_amdgcn_rcpf)
  return __builtin_amdgcn_rcpf(x);
#else
  return 1.0f / x;
#endif
}

__device__ __forceinline__ float fast_tanh(float x) {
#if __has_builtin(__builtin_amdgcn_tanhf)
  return __builtin_amdgcn_tanhf(x);                 // v_tanh_f32 (gfx1250)
#elif __has_builtin(__builtin_amdgcn_tanh_f32)
  return __builtin_amdgcn_tanh_f32(x);
#else
  // tanh(x) = 1 - 2/(e^{2x}+1);  2*log2(e) = 2.885390081777927
  float t = fast_exp2(x * 2.885390081777927f);
  return 1.0f - 2.0f * fast_rcp(t + 1.0f);
#endif
}

// ---------- WMMA f32 16x16x4 fragment helpers ------------------------------
// A (16x4, MxK): lanes 0-15 hold {K=0,K=1}, lanes 16-31 hold {K=2,K=3}; M = lane%16
// B (4x16, KxN): VGPR0: lanes0-15 row K=0, lanes16-31 row K=2; VGPR1: K=1 / K=3; N = lane%16
// C/D (16x16):   VGPR v: lanes0-15 -> M=v, lanes16-31 -> M=v+8; N = lane%16

// ---------------------------------------------------------------------------
// Kernel 1: att_query = query @ W_q^T + b_q      (M=512, N=512, K=512)
//   A[r][d] = query_flat[r][d]     B[d][n] = W_q[n][d]
// ---------------------------------------------------------------------------
__global__ void k_gemm_attquery(const float* __restrict__ Q,
                                const float* __restrict__ Wq,   // [N][D] row-major
                                const float* __restrict__ bq,
                                float* __restrict__ AQ)         // [NROWS][NDIM]
{
  const int tile = blockIdx.x * (blockDim.x >> 5) + (threadIdx.x >> 5);
  const int lane = threadIdx.x & 31;
  const int mt = tile >> 5, nt = tile & 31;      // 32x32 tiles of 16x16
  const int M0 = mt * 16, N0 = nt * 16;
  const int half = lane >> 4;                    // 0: K lo pair, 1: K hi pair
  const int l = lane & 15;

  v8f acc = {};
  const float* arow = Q  + (M0 + l) * NDIM;
  const float* brow = Wq + (N0 + l) * NDIM;      // B[d][n] = Wq[n][d]
  for (int k0 = 0; k0 < NDIM; k0 += 4) {
    v2f a, b;
    a.x = arow[k0 + 2 * half + 0];
    a.y = arow[k0 + 2 * half + 1];
    b.x = brow[k0 + 2 * half + 0];
    b.y = brow[k0 + 2 * half + 1];
    acc = __builtin_amdgcn_wmma_f32_16x16x4_f32(false, a, false, b,
                                                (short)0, acc, false, false);
  }
  const int col = N0 + l;
  const float bias = bq[col];
#pragma unroll
  for (int v = 0; v < 8; ++v) {
    const int row = M0 + v + half * 8;
    AQ[row * NDIM + col] = acc[v] + bias;
  }
}

// ---------------------------------------------------------------------------
// Kernel 2: fused additive-attention scores + softmax.  One block per (b,q).
//   scores[k] = sum_n tanh(aq[n] + keys[b,k,n]) * w_att[n] + b_att
//   probs = softmax_k(scores)  -> written to d_out probs section
// 256 threads = 8 waves; each wave owns 64 k's; each lane owns 16 n's.
// ---------------------------------------------------------------------------
__global__ void k_scores_softmax(const float* __restrict__ AQ,    // [NROWS][NDIM]
                                 const float* __restrict__ keys,  // [B][TK][NDIM]
                                 const float* __restrict__ w_att, // [NDIM]
                                 const float* __restrict__ b_att, // [1]
                                 float* __restrict__ probs)       // [NROWS][TK]
{
  __shared__ float s_sc[TK];
  __shared__ float red_max[8];
  __shared__ float red_sum[8];

  const int row  = blockIdx.x;          // b*TQ + q
  const int b    = row / TQ;
  const int lane = threadIdx.x & 31;
  const int wave = threadIdx.x >> 5;    // 0..7

  // Register-resident att_query row and w_att (16 values per lane, n = lane+32j)
  float aqr[16], wr[16];
  const float* aq_row = AQ + row * NDIM;
#pragma unroll
  for (int j = 0; j < 16; ++j) {
    const int n = lane + 32 * j;
    aqr[j] = aq_row[n];
    wr[j]  = w_att[n];
  }

  const float* kb   = keys + (size_t)b * TK * NDIM;
  const float  batt = b_att[0];

  for (int i = 0; i < 64; ++i) {
    const int k = wave * 64 + i;
    const float* krow = kb + k * NDIM;
    float p = 0.0f;
#pragma unroll
    for (int j = 0; j < 16; ++j) {
      const float t = fast_tanh(aqr[j] + krow[lane + 32 * j]); // coalesced per j
      p = fmaf(t, wr[j], p);
    }
#pragma unroll
    for (int off = 16; off > 0; off >>= 1) p += __shfl_xor(p, off, 32);
    if (lane == 0) s_sc[k] = p + batt;
  }
  __syncthreads();

  // ---- softmax over 512 scores: each thread owns 2 ----
  const int t  = threadIdx.x;
  const float s0 = s_sc[t];
  const float s1 = s_sc[t + 256];
  float m = fmaxf(s0, s1);
#pragma unroll
  for (int off = 16; off > 0; off >>= 1) m = fmaxf(m, __shfl_xor(m, off, 32));
  if (lane == 0) red_max[wave] = m;
  __syncthreads();
  float gm = red_max[0];
#pragma unroll
  for (int i = 1; i < 8; ++i) gm = fmaxf(gm, red_max[i]);

  const float L2E = 1.4426950408889634f;
  const float e0 = fast_exp2((s0 - gm) * L2E);
  const float e1 = fast_exp2((s1 - gm) * L2E);
  float ss = e0 + e1;
#pragma unroll
  for (int off = 16; off > 0; off >>= 1) ss += __shfl_xor(ss, off, 32);
  if (lane == 0) red_sum[wave] = ss;
  __syncthreads();
  float gs = red_sum[0];
#pragma unroll
  for (int i = 1; i < 8; ++i) gs += red_sum[i];
  const float inv = fast_rcp(gs);

  probs[row * TK + t]       = e0 * inv;
  probs[row * TK + t + 256] = e1 * inv;
}

// ---------------------------------------------------------------------------
// Kernel 3: context = probs @ values   (batched: 4 x [128x512]x[512x512])
//   A[q][k] = probs[b][q][k]   B[k][n] = values[b][k][n]  (natural row-major)
// ---------------------------------------------------------------------------
__global__ void k_gemm_context(const float* __restrict__ P,   // [NROWS][TK]
                               const float* __restrict__ V,   // [B][TK][NDIM]
                               float* __restrict__ CTX)       // [NROWS][NDIM]
{
  const int tile = blockIdx.x * (blockDim.x >> 5) + (threadIdx.x >> 5);
  const int lane = threadIdx.x & 31;
  const int b   = tile >> 8;            // 256 tiles per batch (8 x 32)
  const int rem = tile & 255;
  const int mt = rem >> 5, nt = rem & 31;
  const int M0 = mt * 16, N0 = nt * 16;
  const int half = lane >> 4;
  const int l = lane & 15;

  v8f acc = {};
  const float* arow = P + ((size_t)b * TQ + M0 + l) * TK;
  const float* vb   = V + (size_t)b * TK * NDIM;
  for (int k0 = 0; k0 < TK; k0 += 4) {
    v2f a, bb;
    a.x  = arow[k0 + 2 * half + 0];
    a.y  = arow[k0 + 2 * half + 1];
    bb.x = vb[(k0 + 2 * half + 0) * NDIM + N0 + l];   // coalesced across lanes
    bb.y = vb[(k0 + 2 * half + 1) * NDIM + N0 + l];
    acc = __builtin_amdgcn_wmma_f32_16x16x4_f32(false, a, false, bb,
                                                (short)0, acc, false, false);
  }
  const int col = N0 + l;
#pragma unroll
  for (int v = 0; v < 8; ++v) {
    const int row = b * TQ + M0 + v + half * 8;       // flattened (b,q)
    CTX[row * NDIM + col] = acc[v];
  }
}

// ---------------------------------------------------------------------------
// Kernel 4: out = tanh(cat(query, context) @ W_out^T + b_out)
//   Split K=1024 into the query half and the context half of W_out.
// ---------------------------------------------------------------------------
__global__ void k_gemm_out(const float* __restrict__ Q,     // [NROWS][NDIM]
                           const float* __restrict__ CTX,   // [NROWS][NDIM]
                           const float* __restrict__ Wo,    // [NOUT][CATDIM]
                           const float* __restrict__ bo,    // [NOUT]
                           float* __restrict__ OUT)         // [NROWS][NOUT]
{
  const int tile = blockIdx.x * (blockDim.x >> 5) + (threadIdx.x >> 5);
  const int lane = threadIdx.x & 31;
  const int mt = tile >> 5, nt = tile & 31;
  const int M0 = mt * 16, N0 = nt * 16;
  const int half = lane >> 4;
  const int l = lane & 15;

  v8f acc = {};
  const float* qrow = Q   + (M0 + l) * NDIM;
  const float* crow = CTX + (M0 + l) * NDIM;
  const float* wrow = Wo  + (N0 + l) * CATDIM;        // B[c][o] = Wo[o][c]
  for (int k0 = 0; k0 < NDIM; k0 += 4) {              // query half
    v2f a, b;
    a.x = qrow[k0 + 2 * half + 0];
    a.y = qrow[k0 + 2 * half + 1];
    b.x = wrow[k0 + 2 * half + 0];
    b.y = wrow[k0 + 2 * half + 1];
    acc = __builtin_amdgcn_wmma_f32_16x16x4_f32(false, a, false, b,
                                                (short)0, acc, false, false);
  }
  for (int k0 = 0; k0 < NDIM; k0 += 4) {              // context half
    v2f a, b;
    a.x = crow[k0 + 2 * half + 0];
    a.y = crow[k0 + 2 * half + 1];
    b.x = wrow[NDIM + k0 + 2 * half + 0];
    b.y = wrow[NDIM + k0 + 2 * half + 1];
    acc = __builtin_amdgcn_wmma_f32_16x16x4_f32(false, a, false, b,
                                                (short)0, acc, false, false);
  }
  const int col  = N0 + l;
  const float bias = bo[col];
#pragma unroll
  for (int v = 0; v < 8; ++v) {
    const int row = M0 + v + half * 8;
    OUT[row * NOUT + col] = fast_tanh(acc[v] + bias);
  }
}

// ---------------------------------------------------------------------------
extern "C" void kernel_launch(void* const* d_in, const int* in_sizes, int n_in,
                              void* d_out, int out_size, void* d_ws, size_t ws_size,
                              hipStream_t stream) {
  const float* query  = (const float*)d_in[0];   // [4][128][512]
  const float* keys   = (const float*)d_in[1];   // [4][512][512]
  const float* values = (const float*)d_in[2];   // [4][512][512]
  const float* W_q    = (const float*)d_in[3];   // [512][512]
  const float* b_q    = (const float*)d_in[4];   // [512]
  const float* w_att  = (const float*)d_in[5];   // [512]
  const float* b_att  = (const float*)d_in[6];   // [1]
  const float* W_out  = (const float*)d_in[7];   // [512][1024]
  const float* b_out  = (const float*)d_in[8];   // [512]

  float* out   = (float*)d_out;                  // [512][512] (b*t_q, out)
  float* probs = out + (size_t)NROWS * NOUT;     // [512][512] (b*t_q, t_k)

  float* AQ  = (float*)d_ws;                     // [512][512] att_query
  float* CTX = AQ + (size_t)NROWS * NDIM;        // [512][512] context

  // 1) att_query GEMM: 1024 16x16 tiles, 4 waves (128 thr) per block
  k_gemm_attquery<<<256, 128, 0, stream>>>(query, W_q, b_q, AQ);
  // 2) fused scores + softmax, one block per (b,q) row
  k_scores_softmax<<<NROWS, 256, 0, stream>>>(AQ, keys, w_att, b_att, probs);
  // 3) context = probs @ values (batched WMMA GEMM)
  k_gemm_context<<<256, 128, 0, stream>>>(probs, values, CTX);
  // 4) out = tanh([query|context] @ W_out^T + b_out)
  k_gemm_out<<<256, 128, 0, stream>>>(query, CTX, W_out, b_out, out);
}